// FPS_kNN_49331994362179
// MI455X (gfx1250) — compile-verified
//
#include <hip/hip_runtime.h>
#include <hip/hip_bf16.h>
#include <stdint.h>

#define BATCH 4
#define NPTS  8192
#define NSAMP 2048
#define KNN   32
#define CH    128

typedef __attribute__((ext_vector_type(2))) float v2f;
typedef __attribute__((ext_vector_type(8))) float v8f;

// ---------------------------------------------------------------------------
// Kernel 1: Furthest Point Sampling. One workgroup per batch (1024 thr = 32
// waves). Each thread owns 8 points in registers. Per step: distance update,
// packed-u64 argmax via wave shuffles + 32-entry LDS stage.
// Writes fps_idx[B*S] and lc_xyz directly.
// ---------------------------------------------------------------------------
__global__ __launch_bounds__(1024) void fps_kernel(const float* __restrict__ xyz,
                                                   int* __restrict__ fps_idx,
                                                   float* __restrict__ lc_xyz) {
  const int b = blockIdx.x;
  const float* pb = xyz + (size_t)b * NPTS * 3;
  const int t = threadIdx.x;
  const int lane = t & 31, wave = t >> 5;

  float px[8], py[8], pz[8], dist[8];
#pragma unroll
  for (int j = 0; j < 8; ++j) {
    int i = j * 1024 + t;
    px[j] = pb[i * 3 + 0];
    py[j] = pb[i * 3 + 1];
    pz[j] = pb[i * 3 + 2];
    dist[j] = 1e10f;
  }

  __shared__ float cen[3];
  __shared__ unsigned long long red[32];
  __shared__ int sFar;
  if (t == 0) { cen[0] = pb[0]; cen[1] = pb[1]; cen[2] = pb[2]; sFar = 0; }
  __syncthreads();

  for (int s = 0; s < NSAMP; ++s) {
    const float cx = cen[0], cy = cen[1], cz = cen[2];
    if (t == 0) {
      // reference emits the *current* farthest, then updates
      fps_idx[b * NSAMP + s] = sFar;
      size_t o = ((size_t)b * NSAMP + s) * 3;
      lc_xyz[o + 0] = cx; lc_xyz[o + 1] = cy; lc_xyz[o + 2] = cz;
    }
    unsigned long long best = 0ull;
#pragma unroll
    for (int j = 0; j < 8; ++j) {
      float dx = px[j] - cx, dy = py[j] - cy, dz = pz[j] - cz;
      float d = dx * dx + dy * dy + dz * dz;
      dist[j] = fminf(dist[j], d);
      int i = j * 1024 + t;
      // non-negative float bits are order-preserving as uint; complement idx
      // so that ties pick the *lowest* index (matches jnp.argmax).
      unsigned long long key =
          ((unsigned long long)__float_as_uint(dist[j]) << 32) |
          (unsigned)(NPTS - 1 - i);
      best = (key > best) ? key : best;
    }
#pragma unroll
    for (int off = 16; off; off >>= 1) {
      unsigned long long o = __shfl_xor(best, off, 32);
      best = (o > best) ? o : best;
    }
    if (lane == 0) red[wave] = best;
    __syncthreads();
    if (wave == 0) {
      best = red[lane];
#pragma unroll
      for (int off = 16; off; off >>= 1) {
        unsigned long long o = __shfl_xor(best, off, 32);
        best = (o > best) ? o : best;
      }
      if (lane == 0) {
        int i = NPTS - 1 - (int)(best & 0xffffffffu);
        sFar = i;
        cen[0] = pb[i * 3 + 0];
        cen[1] = pb[i * 3 + 1];
        cen[2] = pb[i * 3 + 2];
      }
    }
    __syncthreads();
  }
}

// ---------------------------------------------------------------------------
// Kernel 2: gather lc_x[B,S,128]. One wave per sample, float4 per lane.
// ---------------------------------------------------------------------------
__global__ __launch_bounds__(128) void lcx_gather(const float* __restrict__ x,
                                                  const int* __restrict__ fps_idx,
                                                  float* __restrict__ lc_x) {
  const int w = threadIdx.x >> 5, lane = threadIdx.x & 31;
  const int m = blockIdx.x * 4 + w;          // [0, B*S)
  const int b = m / NSAMP;
  const int idx = fps_idx[m];
  const float4* src = (const float4*)(x + ((size_t)b * NPTS + idx) * CH);
  float4* dst = (float4*)(lc_x + (size_t)m * CH);
  dst[lane] = src[lane];
}

// ---------------------------------------------------------------------------
// Kernel 3: kNN via WMMA f32 16x16x4.
// A (SRC0) = 16 points x K4 = (-2px,-2py,-2pz,|p|^2)
//   lane l<16 holds row M=l  K={0,1};  lane l>=16 holds row M=l-16 K={2,3}
// B (SRC1) = K4 x 16 queries = (qx,qy,qz,1)
//   VGPR v, lane l: K = v + 2*(l>=16), N = l%16
// C = |q_N|^2 broadcast  ->  D[M][N] = squared distance, directly.
// D: lane l holds column N=l%16, rows M = v + 8*(l>=16), v=0..7.
// Point-tile loads are double-buffered (next tile issued before current
// tile's WMMA + heap work) so the 512-iteration loop isn't a serial
// load->wait->use latency chain.
// Each lane keeps a 32-entry LDS max-heap of packed (dist_bits<<32|idx);
// lanes (l, l+16) cover the two point halves; lanes 0..15 merge-select at end.
// ---------------------------------------------------------------------------
__global__ __launch_bounds__(128) void knn_kernel(const float* __restrict__ xyz,
                                                  const float* __restrict__ lc_xyz,
                                                  int* __restrict__ knn_idx) {
  __shared__ unsigned long long heap[4 * 32 * 32];  // [wave][entry][lane]
  const int lane = threadIdx.x & 31, w = threadIdx.x >> 5;
  const int g = blockIdx.x * 4 + w;        // query-tile id, 0..511
  const int b = g >> 7;                    // 128 tiles of 16 queries per batch
  const int s0 = (g & 127) << 4;
  const int qn = lane & 15, hi = lane >> 4;
  const int NT = NPTS / 16;

  unsigned long long* hp = heap + w * 1024 + lane;  // entry e at hp[e*32]
#pragma unroll
  for (int e = 0; e < 32; ++e) hp[e * 32] = ~0ull;
  unsigned long long rootKey = ~0ull;

  const float* q = lc_xyz + ((size_t)b * NSAMP + s0 + qn) * 3;
  const float qx = q[0], qy = q[1], qz = q[2];
  const float sqq = qx * qx + qy * qy + qz * qz;

  v2f bm;
  bm[0] = hi ? qz : qx;
  bm[1] = hi ? 1.0f : qy;
  v8f cin;
#pragma unroll
  for (int v = 0; v < 8; ++v) cin[v] = sqq;

  const float* pbase = xyz + (size_t)b * NPTS * 3;

  // preload tile 0
  float pxv = pbase[qn * 3 + 0];
  float pyv = pbase[qn * 3 + 1];
  float pzv = pbase[qn * 3 + 2];

  for (int t = 0; t < NT; ++t) {
    // issue next tile's load now (clamped, unconditional: no EXEC churn)
    const int tn = (t + 1 < NT) ? (t + 1) : t;
    const float* pn = pbase + (tn * 16 + qn) * 3;
    if (t + 4 < NT)
      __builtin_prefetch(pbase + ((t + 4) * 16 + qn) * 3, 0, 3);
    const float nx = pn[0], ny = pn[1], nz = pn[2];

    v2f am;
    am[0] = hi ? (-2.0f * pzv) : (-2.0f * pxv);
    am[1] = hi ? (pxv * pxv + pyv * pyv + pzv * pzv) : (-2.0f * pyv);

    v8f d = __builtin_amdgcn_wmma_f32_16x16x4_f32(
        /*neg_a=*/false, am, /*neg_b=*/false, bm,
        /*c_mod=*/(short)0, cin, /*reuse_a=*/false, /*reuse_b=*/false);

#pragma unroll
    for (int v = 0; v < 8; ++v) {
      float dd = fmaxf(d[v], 0.0f);  // guard tiny negatives for uint-ordering
      unsigned idxp = (unsigned)(t * 16 + v + 8 * hi);
      unsigned long long key =
          ((unsigned long long)__float_as_uint(dd) << 32) | idxp;
      if (key < rootKey) {           // beats current worst-of-32 -> insert
        int pos = 0;
        for (;;) {                   // sift-down max-heap (depth <= 5)
          int c1 = 2 * pos + 1;
          if (c1 >= 32) break;
          int c2 = c1 + 1;
          unsigned long long k1 = hp[c1 * 32];
          unsigned long long k2 = (c2 < 32) ? hp[c2 * 32] : 0ull;
          int pc = (k2 > k1) ? c2 : c1;
          unsigned long long kc = (k2 > k1) ? k2 : k1;
          if (kc <= key) break;
          hp[pos * 32] = kc;
          pos = pc;
        }
        hp[pos * 32] = key;
        rootKey = hp[0];
      }
    }
    pxv = nx; pyv = ny; pzv = nz;   // rotate double buffer
  }

  // Merge the two half-heaps per query; emit 32 ascending (dist, then idx).
  if (hi == 0) {
    unsigned long long* hp2 = heap + w * 1024 + lane + 16;
    int* out = knn_idx + ((size_t)b * NSAMP + s0 + qn) * KNN;
    for (int r = 0; r < KNN; ++r) {
      unsigned long long bestk = ~0ull;
      int bp = 0;
      for (int e = 0; e < 64; ++e) {
        unsigned long long k = (e < 32) ? hp[e * 32] : hp2[(e - 32) * 32];
        if (k < bestk) { bestk = k; bp = e; }
      }
      if (bp < 32) hp[bp * 32] = ~0ull; else hp2[(bp - 32) * 32] = ~0ull;
      out[r] = (int)(bestk & 0xffffffffu);
    }
  }
}

// ---------------------------------------------------------------------------
// Kernel 4: gather knn_x[B,S,K,128] (one wave per (s,k), float4/lane) and
// knn_xyz[B,S,K,3].
// ---------------------------------------------------------------------------
__global__ __launch_bounds__(256) void knn_gather(const float* __restrict__ xyz,
                                                  const float* __restrict__ x,
                                                  const int* __restrict__ knn_idx,
                                                  float* __restrict__ knn_xyz,
                                                  float* __restrict__ knn_x) {
  const int w = threadIdx.x >> 5, lane = threadIdx.x & 31;
  const size_t g = (size_t)blockIdx.x * 8 + w;  // [0, B*S*K)
  const int b = (int)(g / (NSAMP * KNN));
  const int idx = knn_idx[g];
  const float* row = x + ((size_t)b * NPTS + idx) * CH;
  __builtin_prefetch(row, 0, 0);
  const float4* src = (const float4*)row;
  float4* dst = (float4*)(knn_x + g * CH);
  dst[lane] = src[lane];
  if (lane < 3)
    knn_xyz[g * 3 + lane] = xyz[((size_t)b * NPTS + idx) * 3 + lane];
}

// ---------------------------------------------------------------------------
extern "C" void kernel_launch(void* const* d_in, const int* in_sizes, int n_in,
                              void* d_out, int out_size, void* d_ws, size_t ws_size,
                              hipStream_t stream) {
  const float* xyz = (const float*)d_in[0];  // [4,8192,3]
  const float* x   = (const float*)d_in[1];  // [4,8192,128]

  float* out     = (float*)d_out;
  float* lc_xyz  = out;                       //  4*2048*3      = 24576
  float* lc_x    = out + 24576;               //  4*2048*128    = 1048576
  float* knn_xyz = out + 1073152;             //  4*2048*32*3   = 786432
  float* knn_x   = out + 1859584;             //  4*2048*32*128 = 33554432

  int* fps_idx = (int*)d_ws;                       // 32 KB
  int* knn_idx = (int*)d_ws + BATCH * NSAMP;       // 1 MB

  fps_kernel<<<BATCH, 1024, 0, stream>>>(xyz, fps_idx, lc_xyz);
  lcx_gather<<<(BATCH * NSAMP) / 4, 128, 0, stream>>>(x, fps_idx, lc_x);
  knn_kernel<<<(BATCH * NSAMP / 16) / 4, 128, 0, stream>>>(xyz, lc_xyz, knn_idx);
  knn_gather<<<(BATCH * NSAMP * KNN) / 8, 256, 0, stream>>>(xyz, x, knn_idx,
                                                            knn_xyz, knn_x);
}